// FDoG_90726889161435
// MI455X (gfx1250) — compile-verified
//
#include <hip/hip_runtime.h>

// ---------------------------------------------------------------------------
// FDoG (flow-based DoG) pipeline for MI455X (gfx1250, wave32).
// Stages: WMMA-fp32 Sobel+tangent init -> global max -> normalize ->
//         6x (column-sum + shifted ETF update) -> async-LDS staged DoG ->
//         streamline integration + tanh threshold.
// WMMA kernel: all LDS accesses unconditional; band-matrix constants come
// from compile-time .rodata tables loaded once per lane (no per-chunk
// select chains / divergent EXEC regions).
// ---------------------------------------------------------------------------

typedef __attribute__((ext_vector_type(2))) float v2f;
typedef __attribute__((ext_vector_type(8))) float v8f;

static constexpr int X = 1024, Y = 1024, N = X * Y;

// ---------------- compile-time gaussian weight tables ----------------------
constexpr double cexp(double v) {
  double term = 1.0, sum = 1.0;
  for (int i = 1; i < 80; ++i) { term *= v / (double)i; sum += term; }
  return sum;
}
constexpr double cgauss(double v, double s) {
  return cexp(-(v * v) / (2.0 * s * s)) / (2.5066282746310002 * s);
}

struct DogW_t { float w[19]; float inv_tot; };
constexpr DogW_t make_dogw() {
  DogW_t d{}; double tot = 0.0;
  for (int t = -9; t <= 9; ++t) {
    double w = cgauss((double)t, 3.0) - 0.99 * cgauss((double)t, 4.8);
    d.w[t + 9] = (float)w; tot += w;
  }
  d.inv_tot = (float)(1.0 / tot);
  return d;
}
__device__ constexpr DogW_t DOGW = make_dogw();

struct MW_t { float w[30]; float w0; float inv_tot; };
constexpr MW_t make_mw() {
  MW_t m{}; double s = 0.0;
  for (int i = 1; i <= 30; ++i) { double w = cgauss((double)i, 10.0); m.w[i - 1] = (float)w; s += w; }
  double w0 = cgauss(0.0, 10.0);
  m.w0 = (float)w0; m.inv_tot = (float)(1.0 / (w0 + 2.0 * s));
  return m;
}
__device__ constexpr MW_t MW = make_mw();

// ---------------- compile-time banded Sobel matrices -----------------------
// s = [1,2,1] smoothing, d = [-1,0,1] difference; rows 18,19 zero (K pad).
// BAND.s[q][j] = s[q-j], BAND.d[q][j] = d[q-j]  (zero outside band / q>17).
struct Band_t { float s[20][16]; float d[20][16]; };
constexpr Band_t make_band() {
  Band_t b{};
  for (int q = 0; q < 20; ++q)
    for (int j = 0; j < 16; ++j) {
      int dd = q - j;
      float sv = (dd == 0) ? 1.f : (dd == 1) ? 2.f : (dd == 2) ? 1.f : 0.f;
      float dv = (dd == 0) ? -1.f : (dd == 2) ? 1.f : 0.f;
      b.s[q][j] = (q <= 17) ? sv : 0.f;
      b.d[q][j] = (q <= 17) ? dv : 0.f;
    }
  return b;
}
__device__ constexpr Band_t BAND = make_band();

// ---------------- K0: init global max --------------------------------------
__global__ void init_max_kernel(int* maxv) { *maxv = 0; }

// ---------------- K1: Sobel via fp32 WMMA, tangent init, atomic max --------
// One wave (32 threads) per 16x16 tile.  Gx = Dsel_L * (H * Ssel),
// Gy = Sdel_L * (H * Dsel_R), chained through V_WMMA_F32_16X16X4_F32
// (K chunks of 4, K padded 18->20).
__global__ __launch_bounds__(32) void sobel_wmma_kernel(
    const float* __restrict__ img, float* __restrict__ tang,
    float* __restrict__ magv, int* __restrict__ maxv) {
  __shared__ float H[32][20];       // halo, rows 18..31 / cols 18..19 zero
  __shared__ float SHS_T[16][40];   // (H*Ssel) transposed: [col][row]
  __shared__ float SHD_T[16][40];   // (H*Dsel_R) transposed

  const int b = blockIdx.z;
  const int r0 = blockIdx.y * 16, c0 = blockIdx.x * 16;
  const int lane = threadIdx.x;
  const int half = lane >> 4, l16 = lane & 15;
  const float* src = img + (size_t)b * N;

  // Per-lane constant operand values, loaded once from .rodata.
  // cs/cd[i] = BAND.s/.d[kk][l16] for chunk i (kk = 4i + 2*half).
  float cs0[5], cs1[5], cd0[5], cd1[5];
  #pragma unroll
  for (int i = 0; i < 5; ++i) {
    const int kk = 4 * i + 2 * half;
    cs0[i] = BAND.s[kk][l16];  cs1[i] = BAND.s[kk + 1][l16];
    cd0[i] = BAND.d[kk][l16];  cd1[i] = BAND.d[kk + 1][l16];
  }

  // halo load: clamped address (always valid) + value select, no branches
  for (int idx = lane; idx < 32 * 20; idx += 32) {
    int p = idx / 20, q = idx % 20;
    int gr = r0 - 1 + p, gc = c0 - 1 + q;
    int grc = min(max(gr, 0), X - 1), gcc = min(max(gc, 0), Y - 1);
    float v = src[grc * Y + gcc];
    bool valid = (p < 18) & (q < 18) & (gr >= 0) & (gr < X) & (gc >= 0) & (gc < Y);
    H[p][q] = valid ? v : 0.f;
  }
  __syncthreads();

  // stage 1: HS = H*Ssel (18x16), HD = H*Dsel_R (18x16); two 16-row blocks
  #pragma unroll
  for (int rb = 0; rb < 2; ++rb) {
    v8f hs = {0.f, 0.f, 0.f, 0.f, 0.f, 0.f, 0.f, 0.f};
    v8f hd = {0.f, 0.f, 0.f, 0.f, 0.f, 0.f, 0.f, 0.f};
    const int row = rb * 16 + l16;  // A-matrix row for this lane (< 32)
    #pragma unroll
    for (int i = 0; i < 5; ++i) {
      const int kk = 4 * i + 2 * half;  // <= 19
      v2f a, bs, bd;
      a.x = H[row][kk];
      a.y = H[row][kk + 1];
      bs.x = cs0[i];  bs.y = cs1[i];    // Ssel[kk..][l16]
      bd.x = cd0[i];  bd.y = cd1[i];    // Dsel_R[kk..][l16]
      hs = __builtin_amdgcn_wmma_f32_16x16x4_f32(false, a, false, bs, (short)0, hs, false, false);
      hd = __builtin_amdgcn_wmma_f32_16x16x4_f32(false, a, false, bd, (short)0, hd, false, false);
    }
    const int base = rb * 16 + 8 * half;  // 8 consecutive rows per lane
    #pragma unroll
    for (int g = 0; g < 8; ++g) {
      SHS_T[l16][base + g] = hs[g];
      SHD_T[l16][base + g] = hd[g];
    }
  }
  __syncthreads();

  // stage 2: Gx = Dsel_L*HS, Gy = Sdel_L*HD   (16x18 * 18x16, K padded to 20)
  // A-operand constants reuse the same tables: Dsel_L[m][p] = d[p-m].
  v8f gx = {0.f, 0.f, 0.f, 0.f, 0.f, 0.f, 0.f, 0.f};
  v8f gy = {0.f, 0.f, 0.f, 0.f, 0.f, 0.f, 0.f, 0.f};
  #pragma unroll
  for (int i = 0; i < 5; ++i) {
    const int kk = 4 * i + 2 * half;  // <= 19, rows 18/19 are zero
    v2f ax, ay, bx, by;
    ax.x = cd0[i];  ax.y = cd1[i];            // Dsel_L
    ay.x = cs0[i];  ay.y = cs1[i];            // Sdel_L
    bx.x = SHS_T[l16][kk];  bx.y = SHS_T[l16][kk + 1];  // ds_load_b64
    by.x = SHD_T[l16][kk];  by.y = SHD_T[l16][kk + 1];
    gx = __builtin_amdgcn_wmma_f32_16x16x4_f32(false, ax, false, bx, (short)0, gx, false, false);
    gy = __builtin_amdgcn_wmma_f32_16x16x4_f32(false, ay, false, by, (short)0, gy, false, false);
  }

  // elementwise epilogue in C/D layout: mag, unit tangent (-Gy, Gx)
  float lmax = 0.f;
  #pragma unroll
  for (int g = 0; g < 8; ++g) {
    int r = r0 + g + 8 * half, c = c0 + l16;
    float Gx = gx[g], Gy = gy[g];
    float m = sqrtf(Gx * Gx + Gy * Gy);
    magv[(size_t)b * N + r * Y + c] = m;
    float mm = (m == 0.f) ? 1.f : m;
    tang[(size_t)(b * 2 + 0) * N + r * Y + c] = -Gy / mm;
    tang[(size_t)(b * 2 + 1) * N + r * Y + c] =  Gx / mm;
    lmax = fmaxf(lmax, m);
  }
  #pragma unroll
  for (int off = 16; off; off >>= 1)
    lmax = fmaxf(lmax, __shfl_xor(lmax, off, 32));
  if (lane == 0) atomicMax(maxv, __float_as_int(lmax));  // mag >= 0
}

// ---------------- K2: normalize magnitude by global max --------------------
__global__ void scale_mag_kernel(float* magv, const int* maxv, int n) {
  int i = blockIdx.x * 256 + threadIdx.x;
  if (i >= n) return;
  magv[i] = magv[i] / __int_as_float(*maxv);
}

// ---------------- K3a: dot = sum over x of tang*tang_Y per (b,ch,y) --------
__global__ void colsum_kernel(const float* __restrict__ tang,
                              float* __restrict__ dotv, int ori) {
  const int col = blockIdx.x & (Y - 1);
  const int ch  = (blockIdx.x >> 10) & 1;
  const int b   = blockIdx.x >> 11;
  const float* t = tang + (size_t)(b * 2 + ch) * N;
  float s = 0.f;
  for (int x = threadIdx.x; x < X; x += 256) {
    float a = t[x * Y + col];
    float bv;
    if (ori == 0) { int xs = x + 10;   bv = (xs < X) ? t[xs * Y + col] : 0.f; }
    else          { int ys = col + 10; bv = (ys < Y) ? t[x * Y + ys]  : 0.f; }
    s += a * bv;
  }
  __shared__ float red[256];
  red[threadIdx.x] = s;
  __syncthreads();
  for (int off = 128; off; off >>= 1) {
    if (threadIdx.x < off) red[threadIdx.x] += red[threadIdx.x + off];
    __syncthreads();
  }
  if (threadIdx.x == 0) dotv[blockIdx.x] = red[0];  // (b*2+ch)*Y + col
}

// ---------------- K3b: shifted ETF update + per-pixel renormalize ----------
__global__ void etf_update_kernel(const float* __restrict__ in,
                                  const float* __restrict__ magv,
                                  const float* __restrict__ dotv,
                                  float* __restrict__ out, int ori, int n) {
  int gid = blockIdx.x * 256 + threadIdx.x;
  if (gid >= n) return;
  int b = gid / N, pix = gid % N, x = pix / Y, y = pix % Y;
  float t0, t1, mY;
  if (ori == 0) {
    int xs = x + 10; bool v = xs < X; int s = xs * Y + y;
    t0 = v ? in[(size_t)(b * 2 + 0) * N + s] : 0.f;
    t1 = v ? in[(size_t)(b * 2 + 1) * N + s] : 0.f;
    mY = v ? magv[(size_t)b * N + s] : 0.f;
  } else {
    int ys = y + 10; bool v = ys < Y; int s = x * Y + ys;
    t0 = v ? in[(size_t)(b * 2 + 0) * N + s] : 0.f;
    t1 = v ? in[(size_t)(b * 2 + 1) * N + s] : 0.f;
    mY = v ? magv[(size_t)b * N + s] : 0.f;
  }
  float f = (mY - magv[(size_t)b * N + pix] + 1.f) * 0.5f;
  float d0 = dotv[(b * 2 + 0) * Y + y];
  float d1 = dotv[(b * 2 + 1) * Y + y];
  float n0 = t0 * f * d0, n1 = t1 * f * d1;
  float m = sqrtf(n0 * n0 + n1 * n1);
  if (m == 0.f) m = 1.f;
  out[(size_t)(b * 2 + 0) * N + pix] = n0 / m;
  out[(size_t)(b * 2 + 1) * N + pix] = n1 / m;
}

// ---------------- K4: DoG with async global->LDS halo staging --------------
// 16x16 output tile, 34x34 halo staged via global_load_async_to_lds_b32.
__global__ __launch_bounds__(256) void dog_kernel(
    const float* __restrict__ img, const float* __restrict__ etf,
    float* __restrict__ dogv) {
  __shared__ float T[34][35];
  const int b = blockIdx.z;
  const int c0 = blockIdx.x * 16, r0 = blockIdx.y * 16;
  const int tid = threadIdx.y * 16 + threadIdx.x;
  const float* src = img + (size_t)b * N;
  const unsigned ldsBase = (unsigned)(unsigned long long)&T[0][0];

  for (int idx = tid; idx < 34 * 34; idx += 256) {
    int p = idx / 34, q = idx % 34;
    int gr = min(max(r0 - 9 + p, 0), X - 1);
    int gc = min(max(c0 - 9 + q, 0), Y - 1);
    const float* ga = src + gr * Y + gc;
    unsigned loff = ldsBase + (unsigned)((p * 35 + q) * 4);
    asm volatile("global_load_async_to_lds_b32 %0, %1, off"
                 :: "v"(loff), "v"(ga) : "memory");
  }
  asm volatile("s_wait_asynccnt 0" ::: "memory");
  __syncthreads();

  const int r = r0 + threadIdx.y, c = c0 + threadIdx.x;
  const float e0 = etf[(size_t)(b * 2 + 0) * N + r * Y + c];
  const float e1 = etf[(size_t)(b * 2 + 1) * N + r * Y + c];
  const float p0d = -e1, p1d = e0;  // perpendicular of flow
  float acc = 0.f;
  #pragma unroll
  for (int t = -9; t <= 9; ++t) {
    float px = fminf(fmaxf((float)r + p0d * (float)t, 0.f), (float)(X - 1));
    float py = fminf(fmaxf((float)c + p1d * (float)t, 0.f), (float)(Y - 1));
    int ri = (int)rintf(px), ci = (int)rintf(py);  // round-half-even
    acc += T[ri - (r0 - 9)][ci - (c0 - 9)] * DOGW.w[t + 9];
  }
  dogv[(size_t)b * N + r * Y + c] = acc * DOGW.inv_tot;
}

// ---------------- K5: streamline integration + threshold -------------------
__global__ void fdog_kernel(const float* __restrict__ dogv,
                            const float* __restrict__ etf,
                            int* __restrict__ outp, int n) {
  int gid = blockIdx.x * 256 + threadIdx.x;
  if (gid >= n) return;
  int b = gid / N, pix = gid % N, r = pix / Y, c = pix % Y;
  const float* e0p = etf + (size_t)(b * 2 + 0) * N;
  const float* e1p = etf + (size_t)(b * 2 + 1) * N;
  const float* dp  = dogv + (size_t)b * N;
  float f = dp[pix] * MW.w0;
  #pragma unroll
  for (int dir = 0; dir < 2; ++dir) {
    const float sgn = dir ? 1.f : -1.f;
    float p0 = (float)r, p1 = (float)c;
    int i0 = r, i1 = c;
    float a = 0.f;
    for (int s = 0; s < 30; ++s) {
      int flat = i0 * Y + i1;
      p0 += e0p[flat] * sgn;
      p1 += e1p[flat] * sgn;
      p0 = fminf(fmaxf(p0, 0.f), (float)(X - 1));
      p1 = fminf(fmaxf(p1, 0.f), (float)(Y - 1));
      i0 = (int)rintf(p0); i1 = (int)rintf(p1);
      a += dp[i0 * Y + i1] * MW.w[s];
    }
    f += a;
  }
  f *= MW.inv_tot;
  outp[gid] = ((f < 0.f) && (1.f + tanhf(f) < 0.7f)) ? 0 : 1;
}

// ---------------------------------------------------------------------------
extern "C" void kernel_launch(void* const* d_in, const int* in_sizes, int n_in,
                              void* d_out, int out_size, void* d_ws, size_t ws_size,
                              hipStream_t stream) {
  (void)n_in; (void)out_size; (void)ws_size;
  const float* img = (const float*)d_in[0];
  const int B = in_sizes[0] / N;  // 2

  float* ws    = (float*)d_ws;
  float* tangA = ws;                           // 2*B*N
  float* tangB = tangA + (size_t)2 * B * N;    // 2*B*N
  float* magv  = tangB + (size_t)2 * B * N;    // B*N
  float* dogv  = magv  + (size_t)B * N;        // B*N
  float* dotv  = dogv  + (size_t)B * N;        // B*2*Y
  int*   maxv  = (int*)(dotv + (size_t)B * 2 * Y);

  const int nPix = B * N;
  const int nb = (nPix + 255) / 256;

  init_max_kernel<<<1, 1, 0, stream>>>(maxv);

  dim3 tiles(Y / 16, X / 16, B);
  sobel_wmma_kernel<<<tiles, 32, 0, stream>>>(img, tangA, magv, maxv);
  scale_mag_kernel<<<nb, 256, 0, stream>>>(magv, maxv, nPix);

  for (int p = 0; p < 6; ++p) {                // 3 iterations x (V, H)
    const int ori = p & 1;                     // 0 = V (shift x), 1 = H (shift y)
    const float* tin = (p % 2 == 0) ? tangA : tangB;
    float* tout      = (p % 2 == 0) ? tangB : tangA;
    colsum_kernel<<<B * 2 * Y, 256, 0, stream>>>(tin, dotv, ori);
    etf_update_kernel<<<nb, 256, 0, stream>>>(tin, magv, dotv, tout, ori, nPix);
  }
  // final etf in tangA after 6 passes

  dog_kernel<<<tiles, dim3(16, 16), 0, stream>>>(img, tangA, dogv);
  fdog_kernel<<<nb, 256, 0, stream>>>(dogv, tangA, (int*)d_out, nPix);
}